// GCNModelVAE_62259845923278
// MI455X (gfx1250) — compile-verified
//
#include <hip/hip_runtime.h>
#include <hip/hip_bf16.h>
#include <math.h>

// GCN forward on MI455X (gfx1250, wave32):
//   out = segsum_dst( norm_e * (x @ (W_lin@W_gc1) + b_lin@W_gc1)[src] ) + b_gc1
// xw (12.8MB) and out (12.8MB) stay resident in the 192MB L2, so the edge
// phase (gather + global_atomic_add_f32 scatter) is L2-bound, not HBM-bound.
// The fused GEMM uses V_WMMA_F32_16X16X4_F32 (full fp32 precision).
// W is staged into LDS K-pair-interleaved so every WMMA B fragment is one
// aligned ds_load_b64 (no register repacking in the inner loop).

typedef float v2f __attribute__((ext_vector_type(2)));
typedef float v8f __attribute__((ext_vector_type(8)));

#define D_IN  128
#define H_OUT 32

// ---------- Kernel 1: W_comb = W_lin @ W_gc1 (128x32), b_comb = b_lin @ W_gc1
__global__ __launch_bounds__(128) void k_wcomb(const float* __restrict__ W_lin,
                                               const float* __restrict__ b_lin,
                                               const float* __restrict__ W_gc1,
                                               float* __restrict__ Wc,
                                               float* __restrict__ bc) {
  __shared__ float sG[H_OUT * H_OUT];
  const int t = threadIdx.x;  // 128 threads; thread t owns row k=t of W_comb
  for (int i = t; i < H_OUT * H_OUT; i += 128) sG[i] = W_gc1[i];
  __syncthreads();
  float a[H_OUT];
#pragma unroll
  for (int j = 0; j < H_OUT; ++j) a[j] = W_lin[t * H_OUT + j];
#pragma unroll 4
  for (int h = 0; h < H_OUT; ++h) {
    float acc = 0.f;
#pragma unroll
    for (int j = 0; j < H_OUT; ++j) acc = fmaf(a[j], sG[j * H_OUT + h], acc);
    Wc[t * H_OUT + h] = acc;
  }
  if (t < H_OUT) {
    float acc = 0.f;
#pragma unroll
    for (int j = 0; j < H_OUT; ++j) acc = fmaf(b_lin[j], sG[j * H_OUT + t], acc);
    bc[t] = acc;
  }
}

// ---------- Kernel 2: out[n][h] = b_gc1[h]; zero degree accumulators
__global__ __launch_bounds__(256) void k_init(float* __restrict__ out,
                                              const float* __restrict__ b_gc1,
                                              float* __restrict__ rout,
                                              float* __restrict__ rin,
                                              int n_nodes) {
  const int i = blockIdx.x * blockDim.x + threadIdx.x;
  if (i < n_nodes * H_OUT) out[i] = b_gc1[i & (H_OUT - 1)];
  if (i < n_nodes) { rout[i] = 0.f; rin[i] = 0.f; }
}

// ---------- Kernel 3: degree counting (fp32 atomic add of 1.0, exact < 2^24)
__global__ __launch_bounds__(256) void k_degree(const int* __restrict__ src,
                                                const int* __restrict__ dst,
                                                float* __restrict__ rout,
                                                float* __restrict__ rin,
                                                int n_edges) {
  const int e = blockIdx.x * blockDim.x + threadIdx.x;
  if (e >= n_edges) return;
  atomicAdd(&rout[src[e]], 1.0f);
  atomicAdd(&rin[dst[e]], 1.0f);
}

// ---------- Kernel 4: in-place deg -> rsqrt(max(deg,1))
__global__ __launch_bounds__(256) void k_rinv(float* __restrict__ rout,
                                              float* __restrict__ rin,
                                              int n_nodes) {
  const int i = blockIdx.x * blockDim.x + threadIdx.x;
  if (i >= n_nodes) return;
  rout[i] = rsqrtf(fmaxf(rout[i], 1.0f));
  rin[i]  = rsqrtf(fmaxf(rin[i], 1.0f));
}

// ---------- Kernel 5: xw = x @ Wc + bc via V_WMMA_F32_16X16X4_F32
// One wave (32 lanes) per 16-row node tile; K=128 -> 32 WMMA steps; two v8f
// accumulators cover output columns 0..15 and 16..31.
//
// ISA 7.12.2 fragment layouts (f32, wave32):
//  A 16x4:  lane holds row M=lane%16; VGPR j = K = 2*(lane/16) + j
//  B 4x16:  lane holds col N=lane%16; VGPR j = K = 2*(lane/16) + j
//  C/D 16x16: VGPR v: M = v + 8*(lane/16), N = lane%16
// => both A and B fragments are v2f with two CONSECUTIVE K values, so we
// lay out LDS so each fragment is one aligned 8-byte load:
//  sX : row-major rows of x (K consecutive)           -> A = b64 load
//  sWt: sWt[kp*64 + c*2 + j] = Wc[(2*kp+j)*32 + c]    -> B = b64 load
__global__ __launch_bounds__(128) void k_gemm_wmma(const float* __restrict__ x,
                                                   const float* __restrict__ Wc,
                                                   const float* __restrict__ bc,
                                                   float* __restrict__ xw,
                                                   int n_tiles) {
  __shared__ float sWt[D_IN * H_OUT];  // 16 KB: K-pair-interleaved W_comb
  __shared__ float sX[4][16 * D_IN];   // 32 KB: 4 waves x (16 rows of x)
  const int tid  = threadIdx.x;
  const int wave = tid >> 5;
  const int lane = tid & 31;

  {  // stage Wc with coalesced global reads, interleaved LDS writes
    for (int i = tid; i < D_IN * H_OUT; i += 128) {
      const int k  = i >> 5;        // 0..127 (row of Wc)
      const int c  = i & 31;        // column
      const int kp = k >> 1;        // K-pair index
      const int j  = k & 1;
      sWt[kp * 64 + c * 2 + j] = Wc[i];
    }
  }
  __syncthreads();

  const int tile = blockIdx.x * 4 + wave;
  if (tile >= n_tiles) return;

  float* sx = sX[wave];
  {  // stage 16 rows x 128 f32 of x: 16 float4 per lane, coalesced
    const float4* xg  = (const float4*)(x + (size_t)tile * (16 * D_IN));
    float4*       sx4 = (float4*)sx;
#pragma unroll
    for (int i = 0; i < 16; ++i) sx4[i * 32 + lane] = xg[i * 32 + lane];
  }

  const int r     = lane & 15;   // row (A) / column (B, C/D)
  const int khalf = lane >> 4;   // which K pair of the 4-wide K slab

  const float* aBase  = sx + r * D_IN + khalf * 2;   // + kb*4
  const float* bBase  = sWt + khalf * 64 + r * 2;    // + kb*128
  v8f acc0 = {};  // N = 0..15
  v8f acc1 = {};  // N = 16..31

#pragma unroll 8
  for (int kb = 0; kb < D_IN / 4; ++kb) {
    const v2f a  = *(const v2f*)(aBase + kb * 4);
    const v2f b0 = *(const v2f*)(bBase + kb * 128);        // cols 0..15
    const v2f b1 = *(const v2f*)(bBase + kb * 128 + 32);   // cols 16..31
    acc0 = __builtin_amdgcn_wmma_f32_16x16x4_f32(false, a, false, b0,
                                                 (short)0, acc0, false, false);
    acc1 = __builtin_amdgcn_wmma_f32_16x16x4_f32(false, a, false, b1,
                                                 (short)0, acc1, false, false);
  }

  const float bc0 = bc[r];
  const float bc1 = bc[r + 16];
  float* og = xw + ((size_t)tile * 16 + (size_t)khalf * 8) * H_OUT;
#pragma unroll
  for (int v = 0; v < 8; ++v) {
    og[v * H_OUT + r]      = acc0[v] + bc0;
    og[v * H_OUT + r + 16] = acc1[v] + bc1;
  }
}

// ---------- Kernel 6: edge gather / scale / scatter-add (8 lanes per edge)
// 8 consecutive lanes cover one edge's 32 features -> each 128B gather and
// the 32 atomic adds land on a single L2 line per edge.
__global__ __launch_bounds__(256) void k_edge(const int* __restrict__ src,
                                              const int* __restrict__ dst,
                                              const float* __restrict__ xw,
                                              const float* __restrict__ rout,
                                              const float* __restrict__ rin,
                                              float* __restrict__ out,
                                              int n_edges) {
  const long long t = (long long)blockIdx.x * blockDim.x + threadIdx.x;
  const long long e = t >> 3;
  if (e >= n_edges) return;
  const int f = ((int)t & 7) << 2;
  const int s = src[e];
  const int d = dst[e];
  const float nrm = rout[s] * rin[d];
  const float4 v = *(const float4*)(xw + (size_t)s * H_OUT + f);
  float* o = out + (size_t)d * H_OUT + f;
  atomicAdd(o + 0, v.x * nrm);
  atomicAdd(o + 1, v.y * nrm);
  atomicAdd(o + 2, v.z * nrm);
  atomicAdd(o + 3, v.w * nrm);
}

static inline size_t align256(size_t v) { return (v + 255) & ~(size_t)255; }

extern "C" void kernel_launch(void* const* d_in, const int* in_sizes, int n_in,
                              void* d_out, int out_size, void* d_ws, size_t ws_size,
                              hipStream_t stream) {
  const float* x     = (const float*)d_in[0];
  const int*   src   = (const int*)d_in[1];
  const int*   dst   = (const int*)d_in[2];
  const float* W_lin = (const float*)d_in[3];
  const float* b_lin = (const float*)d_in[4];
  const float* W_gc1 = (const float*)d_in[5];
  const float* b_gc1 = (const float*)d_in[6];
  float* out = (float*)d_out;

  const int n_nodes = in_sizes[0] / D_IN;   // 100000
  const int n_edges = in_sizes[1];          // 3200000

  // workspace layout
  char*  ws = (char*)d_ws;
  size_t o  = 0;
  float* xw   = (float*)(ws + o); o += align256((size_t)n_nodes * H_OUT * sizeof(float));
  float* rout = (float*)(ws + o); o += align256((size_t)n_nodes * sizeof(float));
  float* rin  = (float*)(ws + o); o += align256((size_t)n_nodes * sizeof(float));
  float* Wc   = (float*)(ws + o); o += align256((size_t)D_IN * H_OUT * sizeof(float));
  float* bc   = (float*)(ws + o); o += align256((size_t)H_OUT * sizeof(float));

  k_wcomb<<<1, 128, 0, stream>>>(W_lin, b_lin, W_gc1, Wc, bc);

  const int total_out = n_nodes * H_OUT;
  k_init<<<(total_out + 255) / 256, 256, 0, stream>>>(out, b_gc1, rout, rin, n_nodes);

  k_degree<<<(n_edges + 255) / 256, 256, 0, stream>>>(src, dst, rout, rin, n_edges);

  k_rinv<<<(n_nodes + 255) / 256, 256, 0, stream>>>(rout, rin, n_nodes);

  const int n_tiles = (n_nodes + 15) / 16;  // 6250 (N divisible by 16)
  k_gemm_wmma<<<(n_tiles + 3) / 4, 128, 0, stream>>>(x, Wc, bc, xw, n_tiles);

  const long long edge_threads = (long long)n_edges * 8;
  k_edge<<<(int)((edge_threads + 255) / 256), 256, 0, stream>>>(src, dst, xw,
                                                                rout, rin, out, n_edges);
}